// Atmoshpere_29248727286480
// MI455X (gfx1250) — compile-verified
//
#include <hip/hip_runtime.h>
#include <hip/hip_bf16.h>

typedef float v2f __attribute__((ext_vector_type(2)));
typedef float v8f __attribute__((ext_vector_type(8)));

// ---------------- constants ----------------
namespace {
constexpr int NZ = 40, NY = 384, NX = 512;
constexpr int NTAB = 79;
constexpr float DX = 0.002f, DY = 0.002f, Y0f = 0.1f;
constexpr float DZ = 1.0f / 40.0f;
constexpr float GRAV = 9.8f, PSEA = 101325.0f, KAPPA = 0.286f;
constexpr float RE = 6400000.0f, CPD = 1004.67f, DIFFC = 1.0e9f;
constexpr size_t S2   = (size_t)NY * NX;                 // 196608
constexpr size_t S3   = (size_t)NZ * S2;                 // theta-like
constexpr size_t S3Z1 = (size_t)(NZ + 1) * S2;           // P_/w-like
constexpr size_t SU   = (size_t)NZ * NY * (NX + 1);      // u-like
constexpr size_t SV   = (size_t)NZ * (NY + 1) * NX;      // v-like
}

// standard-atmosphere tables (pre-scaled: HP[m], PP[Pa], TP[K])
__constant__ float c_HP[NTAB] = {
  0.f,100.f,200.f,300.f,400.f,500.f,600.f,700.f,800.f,900.f,1000.f,1500.f,2000.f,2500.f,
  3000.f,3500.f,4000.f,4500.f,5000.f,5500.f,6000.f,6500.f,7000.f,7500.f,8000.f,8500.f,
  9000.f,9500.f,10000.f,11000.f,12000.f,13000.f,14000.f,15000.f,16000.f,17000.f,18000.f,
  19000.f,20000.f,21000.f,22000.f,23000.f,24000.f,25000.f,26000.f,27000.f,28000.f,29000.f,
  30000.f,31000.f,32000.f,33000.f,34000.f,35000.f,36000.f,37000.f,38000.f,39000.f,40000.f,
  41000.f,42000.f,43000.f,44000.f,45000.f,46000.f,47000.f,48000.f,49000.f,50000.f,55000.f,
  60000.f,65000.f,70000.f,75000.f,80000.f,85000.f,90000.f,95000.f,100000.f };
__constant__ float c_PP[NTAB] = {
  101325.f,100120.f,98945.f,97772.f,96611.f,95461.f,94322.f,93194.f,92077.f,90971.f,
  89880.f,84559.f,79500.f,74690.f,70120.f,65780.f,61660.f,57750.f,54050.f,50540.f,
  47220.f,44070.f,41110.f,38300.f,35650.f,33150.f,30800.f,28580.f,26500.f,22700.f,
  19400.f,16580.f,14170.f,12110.f,10350.f,8850.f,7570.f,6470.f,5530.f,4730.f,
  4050.f,3470.f,2970.f,2550.f,2190.f,1880.f,1620.f,1390.f,1200.f,1030.f,
  889.f,767.f,663.f,575.f,499.f,433.f,377.f,329.f,287.f,251.f,
  220.f,193.f,169.f,149.f,131.f,116.f,102.f,90.3f,90.3f,42.5f,
  22.f,10.9f,5.22f,2.39f,1.05f,0.45f,0.18f,0.076f,0.032f };
__constant__ float c_TP[NTAB] = {
  288.15f,287.5f,286.85f,286.2f,285.55f,284.9f,284.25f,283.6f,282.95f,282.3f,
  281.65f,278.4f,275.15f,271.91f,268.66f,265.41f,262.17f,258.92f,255.68f,252.43f,
  249.19f,245.94f,242.7f,239.46f,236.22f,232.97f,229.73f,226.49f,223.25f,216.78f,
  216.65f,216.65f,216.65f,216.65f,216.65f,216.65f,216.65f,216.65f,216.65f,217.58f,
  218.57f,219.57f,220.56f,221.55f,222.54f,223.54f,224.53f,225.52f,226.51f,227.5f,
  228.49f,230.97f,233.74f,236.51f,239.28f,242.05f,244.82f,247.58f,250.35f,253.11f,
  255.88f,258.64f,261.4f,264.16f,266.93f,269.68f,270.65f,270.65f,270.65f,260.77f,
  247.02f,233.29f,219.59f,208.4f,198.64f,188.89f,186.87f,188.42f,195.08f };

// ---------------- table interpolation ----------------
__device__ __forceinline__ int iclamp(int x, int lo, int hi) {
  return x < lo ? lo : (x > hi ? hi : x);
}
__device__ __forceinline__ int wrapN(int i) {
  int m = i % NTAB; return m < 0 ? m + NTAB : m;
}
__device__ int tbl_nearest_desc(float p) {
  int j = 0;
  while (j < NTAB && c_PP[NTAB - 1 - j] < p) ++j;   // searchsorted-left on ascending PP
  int j0 = iclamp(j - 1, 0, NTAB - 1);
  int j1 = iclamp(j, 0, NTAB - 1);
  int ja = (fabsf(p - c_PP[NTAB - 1 - j0]) < fabsf(p - c_PP[NTAB - 1 - j1])) ? j0 : j1;
  return NTAB - 1 - ja;
}
__device__ float hight_from_pressure(float p) {
  int i0 = tbl_nearest_desc(p);
  int i1 = wrapN(i0 + ((p > c_PP[i0]) ? 1 : -1));
  return (c_HP[i1] - c_HP[i0]) * (p - c_PP[i0]) / (c_PP[i1] - c_PP[i0]) + c_HP[i0];
}
__device__ float temperature_from_pressure(float p) {
  int i0 = tbl_nearest_desc(p);
  int i1 = wrapN(i0 + ((p > c_PP[i0]) ? 1 : -1));
  return (c_TP[i1] - c_TP[i0]) * (p - c_PP[i0]) / (c_PP[i1] - c_PP[i0]) + c_TP[i0];
}
__device__ float pressure_from_hight(float z) {
  int j = 0;
  while (j < NTAB && c_HP[j] < z) ++j;
  int j0 = iclamp(j - 1, 0, NTAB - 1);
  int j1 = iclamp(j, 0, NTAB - 1);
  int i0 = (fabsf(z - c_HP[j0]) <= fabsf(z - c_HP[j1])) ? j0 : j1;
  int i1 = wrapN(i0 + ((c_HP[i0] > z) ? 1 : -1));
  return (c_PP[i1] - c_PP[i0]) * (z - c_HP[i0]) / (c_HP[i1] - c_HP[i0]) + c_PP[i0];
}

// ---------------- zero-padded field accessors ----------------
__device__ __forceinline__ float gFx(const float* a, int k, int j, int i) {
  return (j >= 0 && j < NY && i >= 0 && i <= NX) ? a[((size_t)k * NY + j) * (NX + 1) + i] : 0.f;
}
__device__ __forceinline__ float gGy(const float* a, int k, int j, int i) {
  return (j >= 0 && j <= NY && i >= 0 && i < NX) ? a[((size_t)k * (NY + 1) + j) * NX + i] : 0.f;
}
__device__ __forceinline__ float gU(const float* a, int k, int j, int i) {
  return (k >= 0 && k < NZ && j >= 0 && j < NY && i >= 0 && i <= NX)
       ? a[((size_t)k * NY + j) * (NX + 1) + i] : 0.f;
}
__device__ __forceinline__ float gV(const float* a, int k, int j, int i) {
  return (k >= 0 && k < NZ && j >= 0 && j <= NY && i >= 0 && i < NX)
       ? a[((size_t)k * (NY + 1) + j) * NX + i] : 0.f;
}
__device__ __forceinline__ float g2(const float* a, int j, int i) {
  return (j >= 0 && j < NY && i >= 0 && i < NX) ? a[(size_t)j * NX + i] : 0.f;
}
__device__ __forceinline__ float g3(const float* a, int k, int j, int i) {
  return (k >= 0 && k < NZ && j >= 0 && j < NY && i >= 0 && i < NX)
       ? a[((size_t)k * NY + j) * NX + i] : 0.f;
}
__device__ __forceinline__ float gW(const float* a, int k, int j, int i) { // (NZ+1,NY,NX), k in range
  return (j >= 0 && j < NY && i >= 0 && i < NX) ? a[((size_t)k * NY + j) * NX + i] : 0.f;
}

// Arakawa coefficient stencils (zero-padded algebra matches reference exactly)
__device__ __forceinline__ float fB(const float* Fx, int k, int j, int ii) {
  return (gFx(Fx,k,j+1,ii+1)+gFx(Fx,k,j+1,ii) + 2.f*(gFx(Fx,k,j,ii+1)+gFx(Fx,k,j,ii))
        + gFx(Fx,k,j-1,ii+1)+gFx(Fx,k,j-1,ii)) * (1.f/12.f);
}
__device__ __forceinline__ float fC(const float* Gy, int k, int jj, int i) {
  return (gGy(Gy,k,jj+1,i)+gGy(Gy,k,jj+1,i-1) + 2.f*(gGy(Gy,k,jj,i)+gGy(Gy,k,jj,i-1))
        + gGy(Gy,k,jj-1,i)+gGy(Gy,k,jj-1,i-1)) * (1.f/12.f);
}
__device__ __forceinline__ float fDE(const float* Fx, const float* Gy, int k, int jj, int ii, float sgn) {
  float d1 = gGy(Gy,k,jj+1,ii) + 2.f*gGy(Gy,k,jj,ii) + gGy(Gy,k,jj-1,ii);
  float d2 = gFx(Fx,k,jj,ii+1)+gFx(Fx,k,jj,ii)+gFx(Fx,k,jj-1,ii+1)+gFx(Fx,k,jj-1,ii);
  return d1 + sgn * d2 * (1.f/24.f);
}
__device__ __forceinline__ float fQ(const float* Fx, int k, int j, int ii) {
  return (gFx(Fx,k,j,ii-1)+gFx(Fx,k,j-1,ii-1) + 2.f*(gFx(Fx,k,j,ii)+gFx(Fx,k,j-1,ii))
        + gFx(Fx,k,j,ii+1)+gFx(Fx,k,j-1,ii+1)) * (1.f/12.f);
}
__device__ __forceinline__ float fR(const float* Gy, int k, int j, int ii) {
  return (gGy(Gy,k,j+1,ii-1)+gGy(Gy,k,j,ii-1) + 2.f*(gGy(Gy,k,j+1,ii)+gGy(Gy,k,j,ii))
        + gGy(Gy,k,j+1,ii+1)+gGy(Gy,k,j,ii+1)) * (1.f/12.f);
}
__device__ __forceinline__ float fST(const float* Fx, const float* Gy, int k, int j, int ii, float sgn) {
  float s1 = gGy(Gy,k,j+1,ii)+gGy(Gy,k,j+1,ii-1)+gGy(Gy,k,j,ii)+gGy(Gy,k,j,ii-1);
  float s2 = gFx(Fx,k,j,ii+1) + 2.f*gFx(Fx,k,j,ii) + gFx(Fx,k,j,ii-1);
  return s1 + sgn * s2 * (1.f/24.f);
}

// ---------------- init kernels ----------------
__global__ void k_profiles(const int* __restrict__ ptp, float* __restrict__ sigma,
                           float* __restrict__ deltaA) {
  int t = threadIdx.x;
  float pt = (float)ptp[0];
  if (t <= NZ) {
    float zt = hight_from_pressure(pt);
    float z = (1.0f - (float)t / (float)NZ) * zt;
    float pz = pressure_from_hight(z);
    float psrf = pressure_from_hight(0.0f);
    sigma[t] = (pz - pt) / (psrf - pt);
  }
  if (t < NY) {
    deltaA[t] = RE * RE * cosf(Y0f + ((float)t + 0.5f) * DY) * DX * DY;
  }
}

__global__ void k_pi(const float* __restrict__ ps, const int* __restrict__ ptp,
                     float* __restrict__ pi) {
  size_t p = (size_t)blockIdx.x * blockDim.x + threadIdx.x;
  if (p < S2) pi[p] = ps[p] - (float)ptp[0];
}

__global__ void k_initf(const float* __restrict__ pi, const float* __restrict__ sigma,
                        const int* __restrict__ ptp, float* __restrict__ P,
                        float* __restrict__ Pb, float* __restrict__ theta) {
  size_t idx = (size_t)blockIdx.x * blockDim.x + threadIdx.x;
  if (idx >= S3Z1) return;
  int k = (int)(idx / S2);
  size_t p = idx % S2;
  float pt = (float)ptp[0];
  float piv = pi[p];
  float p_k = pt + sigma[k] * piv;
  float Pbk = powf(p_k / PSEA, KAPPA);
  Pb[idx] = Pbk;
  if (k < NZ) {
    float p_k1 = pt + sigma[k + 1] * piv;
    float Pbk1 = powf(p_k1 / PSEA, KAPPA);
    float Pv = (p_k1 * Pbk1 - p_k * Pbk) / (p_k1 - p_k) / (1.f + KAPPA);
    P[idx] = Pv;
    float pres = PSEA * powf(Pv, 1.f / KAPPA);
    theta[idx] = temperature_from_pressure(pres) / Pv;
  }
}

// ---------------- step kernels ----------------
__global__ void k_fx(const float* __restrict__ pi, const float* __restrict__ u,
                     float* __restrict__ Fx) {
  size_t idx = (size_t)blockIdx.x * blockDim.x + threadIdx.x;
  if (idx >= SU) return;
  int i = (int)(idx % (NX + 1));
  int j = (int)((idx / (NX + 1)) % NY);
  float pl = (i - 1 >= 0) ? pi[(size_t)j * NX + i - 1] : 0.f;
  float pr = (i < NX) ? pi[(size_t)j * NX + i] : 0.f;
  Fx[idx] = (pl + pr) * 0.5f * u[idx] * RE * DY;
}

__global__ void k_gy(const float* __restrict__ pi, const float* __restrict__ v,
                     float* __restrict__ Gy) {
  size_t idx = (size_t)blockIdx.x * blockDim.x + threadIdx.x;
  if (idx >= SV) return;
  int i = (int)(idx % NX);
  int j = (int)((idx / NX) % (NY + 1));
  float pd = (j - 1 >= 0) ? pi[(size_t)(j - 1) * NX + i] : 0.f;
  float pu = (j < NY) ? pi[(size_t)j * NX + i] : 0.f;
  Gy[idx] = (pd + pu) * 0.5f * v[idx] * RE * DX * cosf(Y0f + (float)j * DY);
}

// Vertical scan on the matrix pipe: cumsum(div) over 40 levels == lower-triangular
// ones-matrix (48x40 padded) x div(40 x 16px), tiled as V_WMMA_F32_16X16X4_F32.
__global__ void k_column(const float* __restrict__ Fx, const float* __restrict__ Gy,
                         const float* __restrict__ pi0, const float* __restrict__ deltaA,
                         const float* __restrict__ sigma, const int* __restrict__ dtp,
                         float dtscale, float* __restrict__ pi1, float* __restrict__ wout) {
  __shared__ float s_div[8][NZ][16];
  int l = threadIdx.x & 31;
  int wid = threadIdx.x >> 5;
  int gw = blockIdx.x * 8 + wid;
  int p0 = gw * 16;
  float dtf = dtscale * (float)dtp[0];

  for (int t = l; t < NZ * 16; t += 32) {       // stage div tile in LDS
    int k = t >> 4, n = t & 15;
    int p = p0 + n, j = p / NX, i = p % NX;
    float dfx = Fx[((size_t)k * NY + j) * (NX + 1) + i + 1]
              - Fx[((size_t)k * NY + j) * (NX + 1) + i];
    float dgy = Gy[((size_t)k * (NY + 1) + j + 1) * NX + i]
              - Gy[((size_t)k * (NY + 1) + j) * NX + i];
    s_div[wid][k][n] = (dfx + dgy) * DZ;
  }
  __syncthreads();

  int ln = l & 15, nh = l >> 4;
  v8f acc[3];
  #pragma unroll
  for (int r = 0; r < 3; ++r) {
    v8f c = {};
    int kcmax = (r == 2) ? 10 : 4 * (r + 1);
    for (int kc = 0; kc < kcmax; ++kc) {
      int K0 = kc * 4 + 2 * nh;                 // A/B VGPR-lane layout per ISA 7.12.2
      v2f a, b;
      a.x = ((16 * r + ln) >= K0)     ? 1.f : 0.f;   // lower-triangular scan matrix
      a.y = ((16 * r + ln) >= K0 + 1) ? 1.f : 0.f;
      b.x = s_div[wid][K0][ln];
      b.y = s_div[wid][K0 + 1][ln];
      c = __builtin_amdgcn_wmma_f32_16x16x4_f32(false, a, false, b, (short)0, c, false, false);
    }
    acc[r] = c;
  }

  float s39 = __shfl(acc[2][7], ln);            // column total (k=39 row) to all lanes
  int p = p0 + ln, j = p / NX;
  float dA = deltaA[j];
  float pi0v = pi0[p];
  float pi1v = pi0v - dtf * s39 / dA;
  if (l < 16) { pi1[p] = pi1v; wout[p] = 0.f; }
  #pragma unroll
  for (int r = 0; r < 3; ++r) {
    #pragma unroll
    for (int d = 0; d < 8; ++d) {
      int kk = 16 * r + d + 8 * nh;
      if (kk < NZ) {
        float S = acc[r][d];
        float wb = -S / (dA * pi1v) - sigma[kk + 1] * (pi1v - pi0v) / (dtf * pi1v);
        wout[(size_t)(kk + 1) * S2 + p] = wb;
      }
    }
  }
}

__global__ void k_theta(const float* __restrict__ theta, const float* __restrict__ theta0,
                        const float* __restrict__ Fx, const float* __restrict__ Gy,
                        const float* __restrict__ w, const float* __restrict__ P,
                        const float* __restrict__ Pb, const float* __restrict__ pi,
                        const float* __restrict__ pi0, const float* __restrict__ pi1,
                        const float* __restrict__ deltaA, const int* __restrict__ dtp,
                        float dtscale, float* __restrict__ theta1) {
  size_t idx = (size_t)blockIdx.x * blockDim.x + threadIdx.x;
  if (idx >= S3) return;
  __builtin_prefetch(&theta0[idx], 0, 1);       // global_prefetch_b8
  int i = (int)(idx % NX);
  int j = (int)((idx / NX) % NY);
  int k = (int)(idx / S2);
  size_t p = (size_t)j * NX + i;
  float dtf = dtscale * (float)dtp[0];

  float th = theta[idx];
  float thxm = g3(theta,k,j,i-1), thxp = g3(theta,k,j,i+1);
  float thym = g3(theta,k,j-1,i), thyp = g3(theta,k,j+1,i);
  float thzm = g3(theta,k-1,j,i), thzp = g3(theta,k+1,j,i);
  float lap = thxm + thxp + thym + thyp + thzm + thzp - 6.f * th;

  float FxR = Fx[((size_t)k*NY+j)*(NX+1) + i + 1];
  float FxL = Fx[((size_t)k*NY+j)*(NX+1) + i];
  float GyU = Gy[((size_t)k*(NY+1)+j+1)*NX + i];
  float GyD = Gy[((size_t)k*(NY+1)+j)*NX + i];
  float dflux = FxR*(thxp+th) - FxL*(th+thxm) + GyU*(thyp+th) - GyD*(th+thym);

  float thh_k = 0.f, thh_k1 = 0.f;
  if (k >= 1) {
    float Pk = P[idx], Pkm = P[idx - S2];
    thh_k = (Pk*th + Pkm*thzm - (th - thzm)*Pb[idx]) / (Pk - Pkm);
  }
  if (k + 1 <= NZ - 1) {
    float Pk = P[idx], Pkp = P[idx + S2];
    thh_k1 = (Pkp*thzp + Pk*th - (thzp - th)*Pb[idx + S2]) / (Pkp - Pk);
  }
  float vert = w[idx + S2] * thh_k1 - w[idx] * thh_k;

  float dA = deltaA[j];
  float pi1v = pi1[p], pi0v = pi0[p], piv = pi[p];
  theta1[idx] = pi0v / pi1v * theta0[idx]
              + dtf / (dA * pi1v) * (0.5f * dflux + piv * dA * vert / DZ + DIFFC * lap);
}

__global__ void k_phi(const float* __restrict__ P, const float* __restrict__ Pb,
                      const float* __restrict__ theta, const float* __restrict__ zs,
                      const float* __restrict__ sigma, float* __restrict__ phi,
                      float* __restrict__ pgrad) {
  size_t p = (size_t)blockIdx.x * blockDim.x + threadIdx.x;
  if (p >= S2) return;
  float zsv = zs[p];
  float sc = 0.f, se = 0.f;
  for (int k = NZ - 1; k >= 0; --k) {           // reverse scan over 40 levels
    size_t id = (size_t)k * S2 + p;
    float Pk = P[id], th = theta[id];
    float Pbk = Pb[id], Pbk1 = Pb[id + S2];
    float c = -CPD * (Pk - Pbk1) * th;
    float e = -CPD * (Pbk - Pk) * th;
    sc += c; se += e;
    phi[id] = GRAV * zsv + sc + (se - e);
    float sg = sigma[k], sg1 = sigma[k + 1];
    pgrad[id] = th / DZ * ((sg1 * Pbk1 - sg * Pbk) - Pk * (sg1 - sg));
  }
}

__global__ void k_u(const float* __restrict__ Fx, const float* __restrict__ Gy,
                    const float* __restrict__ u, const float* __restrict__ u0,
                    const float* __restrict__ pi, const float* __restrict__ pi0,
                    const float* __restrict__ pi1, const float* __restrict__ w,
                    const float* __restrict__ phi, const float* __restrict__ pgrad,
                    const float* __restrict__ deltaA, const int* __restrict__ dtp,
                    float dtscale, float* __restrict__ u1) {
  size_t idx = (size_t)blockIdx.x * blockDim.x + threadIdx.x;
  if (idx >= SU) return;
  __builtin_prefetch(&u0[idx], 0, 1);
  int i = (int)(idx % (NX + 1));
  int j = (int)((idx / (NX + 1)) % NY);
  int k = (int)(idx / ((size_t)(NX + 1) * NY));
  float dtf = dtscale * (float)dtp[0];

  auto BXU = [&](int ii) -> float {
    if (ii < 0 || ii >= NX) return 0.f;
    return fB(Fx, k, j, ii) * (gU(u, k, j, ii + 1) + gU(u, k, j, ii));
  };
  auto CYU = [&](int jj) -> float {
    return fC(Gy, k, jj, i) * (gU(u, k, jj, i) + gU(u, k, jj - 1, i));
  };
  auto XDE = [&](int jj, int ii, float sgn) -> float {
    if (ii < 0 || ii >= NX) return 0.f;
    float bxy = gU(u,k,jj,ii+1)+gU(u,k,jj-1,ii+1)+gU(u,k,jj,ii)+gU(u,k,jj-1,ii);
    return fDE(Fx, Gy, k, jj, ii, sgn) * bxy;
  };
  float advec = 0.5f * (-(BXU(i) - BXU(i - 1)) - (CYU(j + 1) - CYU(j))
                        + (XDE(j + 1, i, 1.f) - XDE(j, i - 1, 1.f))
                        + (XDE(j, i, -1.f) - XDE(j + 1, i - 1, -1.f)));

  auto Wf = [&](int kk, int jj, int ii) -> float {
    if (jj < 0 || jj >= NY || ii < 0 || ii >= NX) return 0.f;
    return pi[(size_t)jj*NX+ii] * deltaA[jj] * w[((size_t)kk*NY+jj)*NX+ii];
  };
  auto pwA = [&](int kk) -> float {
    return (Wf(kk,j+1,i)+Wf(kk,j+1,i-1) + 2.f*(Wf(kk,j,i)+Wf(kk,j,i-1))
          + Wf(kk,j-1,i)+Wf(kk,j-1,i-1)) * 0.125f;
  };
  float uzm = gU(u, k - 1, j, i), uc = u[idx], uzp = gU(u, k + 1, j, i);
  float trans = -0.5f * (pwA(k + 1) * (uzp + uc) - pwA(k) * (uc + uzm)) / DZ;

  float dphi = g3(phi, k, j, i) - g3(phi, k, j, i - 1);
  float dpi  = g2(pi, j, i) - g2(pi, j, i - 1);
  float bpg  = g3(pgrad, k, j, i) + g3(pgrad, k, j, i - 1);
  float press = -RE * DY * (dphi * dpi * 0.5f + dpi * 0.5f * CPD * bpg);

  float diff = DIFFC * (gU(u,k,j,i-1)+gU(u,k,j,i+1)+gU(u,k,j-1,i)+gU(u,k,j+1,i)
                      + uzm + uzp - 6.f * uc);

  auto pA = [&](const float* pf) -> float {
    auto A0 = [&](int jj, int ii) -> float {
      if (jj < 0 || jj >= NY || ii < 0 || ii >= NX) return 0.f;
      return pf[(size_t)jj*NX+ii] * deltaA[jj];
    };
    return (A0(j+1,i)+A0(j+1,i-1) + 2.f*(A0(j,i)+A0(j,i-1)) + A0(j-1,i)+A0(j-1,i-1)) * 0.125f;
  };
  float p0A = pA(pi0), p1A = pA(pi1);
  u1[idx] = (p0A * u0[idx] + dtf * (advec + trans + press + diff)) / p1A;
}

__global__ void k_v(const float* __restrict__ Fx, const float* __restrict__ Gy,
                    const float* __restrict__ v, const float* __restrict__ v0,
                    const float* __restrict__ pi, const float* __restrict__ pi0,
                    const float* __restrict__ pi1, const float* __restrict__ w,
                    const float* __restrict__ phi, const float* __restrict__ pgrad,
                    const float* __restrict__ deltaA, const int* __restrict__ dtp,
                    float dtscale, float* __restrict__ v1) {
  size_t idx = (size_t)blockIdx.x * blockDim.x + threadIdx.x;
  if (idx >= SV) return;
  __builtin_prefetch(&v0[idx], 0, 1);
  int i = (int)(idx % NX);
  int j = (int)((idx / NX) % (NY + 1));
  int k = (int)(idx / ((size_t)NX * (NY + 1)));
  float dtf = dtscale * (float)dtp[0];

  auto XQ = [&](int ii) -> float {
    return fQ(Fx, k, j, ii) * (gV(v, k, j, ii) + gV(v, k, j, ii - 1));
  };
  auto YR = [&](int jj) -> float {
    if (jj < 0 || jj >= NY) return 0.f;
    return fR(Gy, k, jj, i) * (gV(v, k, jj + 1, i) + gV(v, k, jj, i));
  };
  auto XST = [&](int jj, int ii, float sgn) -> float {
    if (jj < 0 || jj >= NY) return 0.f;
    float bxy = gV(v,k,jj+1,ii)+gV(v,k,jj+1,ii-1)+gV(v,k,jj,ii)+gV(v,k,jj,ii-1);
    return fST(Fx, Gy, k, jj, ii, sgn) * bxy;
  };
  float advec = 0.5f * (-(XQ(i + 1) - XQ(i)) - (YR(j) - YR(j - 1))
                        + (XST(j, i + 1, 1.f) - XST(j - 1, i, 1.f))
                        + (XST(j - 1, i + 1, -1.f) - XST(j, i, -1.f)));

  auto Wf = [&](int kk, int jj, int ii) -> float {
    if (jj < 0 || jj >= NY || ii < 0 || ii >= NX) return 0.f;
    return pi[(size_t)jj*NX+ii] * deltaA[jj] * w[((size_t)kk*NY+jj)*NX+ii];
  };
  auto pwA = [&](int kk) -> float {
    return (Wf(kk,j,i-1)+Wf(kk,j-1,i-1) + 2.f*(Wf(kk,j,i)+Wf(kk,j-1,i))
          + Wf(kk,j,i+1)+Wf(kk,j-1,i+1)) * 0.125f;
  };
  float vzm = gV(v, k - 1, j, i), vc = v[idx], vzp = gV(v, k + 1, j, i);
  float trans = -0.5f * (pwA(k + 1) * (vzp + vc) - pwA(k) * (vc + vzm)) / DZ;

  float dphi = g3(phi, k, j, i) - g3(phi, k, j - 1, i);
  float dpi  = g2(pi, j, i) - g2(pi, j - 1, i);
  float bpg  = g3(pgrad, k, j, i) + g3(pgrad, k, j - 1, i);
  float press = -RE * DX * (dphi * dpi * 0.5f + dpi * 0.5f * CPD * bpg);

  float diff = DIFFC * (gV(v,k,j,i-1)+gV(v,k,j,i+1)+gV(v,k,j-1,i)+gV(v,k,j+1,i)
                      + vzm + vzp - 6.f * vc);

  auto pA = [&](const float* pf) -> float {
    auto A0 = [&](int jj, int ii) -> float {
      if (jj < 0 || jj >= NY || ii < 0 || ii >= NX) return 0.f;
      return pf[(size_t)jj*NX+ii] * deltaA[jj];
    };
    return (A0(j,i-1)+A0(j-1,i-1) + 2.f*(A0(j,i)+A0(j-1,i)) + A0(j,i+1)+A0(j-1,i+1)) * 0.125f;
  };
  float p0A = pA(pi0), p1A = pA(pi1);
  v1[idx] = (p0A * v0[idx] + dtf * (advec + trans + press + diff)) / p1A;
}

// ---------------- host driver ----------------
static void run_step(const float* pi_c, const float* th_c, const float* u_c, const float* v_c,
                     const float* pi_b, const float* th_b, const float* u_b, const float* v_b,
                     float dtscale, const int* dtp, const float* zs,
                     const float* deltaA, const float* sigma,
                     const float* P, const float* Pb,
                     float* Fx, float* Gy, float* ww, float* phi, float* pgrad,
                     float* pi1, float* th1, float* u1, float* v1, hipStream_t stream) {
  dim3 B(256);
  k_fx<<<dim3((unsigned)((SU + 255) / 256)), B, 0, stream>>>(pi_c, u_c, Fx);
  k_gy<<<dim3((unsigned)((SV + 255) / 256)), B, 0, stream>>>(pi_c, v_c, Gy);
  k_column<<<dim3((unsigned)(S2 / 128)), B, 0, stream>>>(Fx, Gy, pi_b, deltaA, sigma, dtp,
                                                         dtscale, pi1, ww);
  k_theta<<<dim3((unsigned)((S3 + 255) / 256)), B, 0, stream>>>(th_c, th_b, Fx, Gy, ww, P, Pb,
                                                                pi_c, pi_b, pi1, deltaA, dtp,
                                                                dtscale, th1);
  k_phi<<<dim3((unsigned)((S2 + 255) / 256)), B, 0, stream>>>(P, Pb, th_c, zs, sigma, phi, pgrad);
  k_u<<<dim3((unsigned)((SU + 255) / 256)), B, 0, stream>>>(Fx, Gy, u_c, u_b, pi_c, pi_b, pi1,
                                                            ww, phi, pgrad, deltaA, dtp,
                                                            dtscale, u1);
  k_v<<<dim3((unsigned)((SV + 255) / 256)), B, 0, stream>>>(Fx, Gy, v_c, v_b, pi_c, pi_b, pi1,
                                                            ww, phi, pgrad, deltaA, dtp,
                                                            dtscale, v1);
}

extern "C" void kernel_launch(void* const* d_in, const int* in_sizes, int n_in,
                              void* d_out, int out_size, void* d_ws, size_t ws_size,
                              hipStream_t stream) {
  (void)in_sizes; (void)n_in; (void)out_size; (void)ws_size;
  const float* ps = (const float*)d_in[0];
  const float* zs = (const float*)d_in[1];
  const float* u  = (const float*)d_in[2];
  const float* v  = (const float*)d_in[3];
  const int* dtp  = (const int*)d_in[4];
  const int* ptp  = (const int*)d_in[5];

  float* w0 = (float*)d_ws;
  size_t off = 0;
  float* sigma  = w0 + off; off += 64;
  float* deltaA = w0 + off; off += 512;
  float* pi     = w0 + off; off += S2;
  float* theta  = w0 + off; off += S3;
  float* P      = w0 + off; off += S3;
  float* Pb     = w0 + off; off += S3Z1;
  float* ph     = w0 + off; off += S2;
  float* th     = w0 + off; off += S3;
  float* uh     = w0 + off; off += SU;
  float* vh     = w0 + off; off += SV;
  float* Fx     = w0 + off; off += SU;
  float* Gy     = w0 + off; off += SV;
  float* ww     = w0 + off; off += S3Z1;
  float* phi    = w0 + off; off += S3;
  float* pgrad  = w0 + off; off += S3;

  float* o_pi1 = (float*)d_out;
  float* o_th1 = o_pi1 + S2;
  float* o_u1  = o_th1 + S3;
  float* o_v1  = o_u1 + SU;

  k_profiles<<<dim3(1), dim3(512), 0, stream>>>(ptp, sigma, deltaA);
  k_pi<<<dim3((unsigned)((S2 + 255) / 256)), dim3(256), 0, stream>>>(ps, ptp, pi);
  k_initf<<<dim3((unsigned)((S3Z1 + 255) / 256)), dim3(256), 0, stream>>>(pi, sigma, ptp,
                                                                          P, Pb, theta);
  // half step: current == base == initial state
  run_step(pi, theta, u, v, pi, theta, u, v, 0.5f, dtp, zs, deltaA, sigma, P, Pb,
           Fx, Gy, ww, phi, pgrad, ph, th, uh, vh, stream);
  // full step: current = half-step state, base = initial state, outputs -> d_out
  run_step(ph, th, uh, vh, pi, theta, u, v, 1.0f, dtp, zs, deltaA, sigma, P, Pb,
           Fx, Gy, ww, phi, pgrad, o_pi1, o_th1, o_u1, o_v1, stream);
}